// DistancePenalty_84731114815830
// MI455X (gfx1250) — compile-verified
//
#include <hip/hip_runtime.h>

typedef float v2f __attribute__((ext_vector_type(2)));
typedef float v8f __attribute__((ext_vector_type(8)));

#define THRESH_MIN 0.9f
#define THRESH_MAX 2.0f
#define FIXED_SCALE 16777216.0   // 2^24 fixed-point scale for deterministic reduce

// Raw v_sqrt_f32 (~1 ulp on normal inputs) — avoids the IEEE-correct
// rescale+refinement expansion that dominated the epilogue VALU count.
__device__ __forceinline__ float fast_sqrtf(float x) {
    return __builtin_amdgcn_sqrtf(x);
}

// ---- zero the u64 fixed-point accumulator (d_ws is not re-poisoned) ----
__global__ void dp_init_kernel(unsigned long long* acc) {
    if (threadIdx.x == 0) acc[0] = 0ull;
}

// ---- main: one wave per (row-tile, 8-wide column-tile strip) ----
// G-tile = A(16x4) x B(4x16) via v_wmma_f32_16x16x4_f32; sq = ni + nj - 2g.
__global__ __launch_bounds__(32) void dp_pairs_kernel(
    const float* __restrict__ geom,          // [3*N] packed xyz
    unsigned long long* __restrict__ acc,
    int nTiles) {
    const int ti     = blockIdx.y;           // row tile (i side)
    const int tjBase = blockIdx.x * 8;       // first column tile of strip
    if (tjBase + 7 < ti) return;             // strip entirely below diagonal

    const int lane = threadIdx.x;            // wave32
    const int la   = lane & 15;
    const int hi   = lane >> 4;              // 0: K={0,1}, 1: K={2,pad}

    // --- A fragment: atom ti*16+la, coords {x,y} (lanes 0-15) or {z,0} (16-31)
    const int ai = ti * 16 + la;
    v2f afrag;
    if (hi == 0) {
        afrag[0] = geom[3 * ai + 0];
        afrag[1] = geom[3 * ai + 1];
    } else {
        afrag[0] = geom[3 * ai + 2];
        afrag[1] = 0.0f;
    }
    // partial norm this lane contributes: x^2+y^2 (lo half) or z^2 (hi half)
    const float sA = afrag[0] * afrag[0] + afrag[1] * afrag[1];

    // ni[r] = ||atom ti*16 + m||^2 with m = r + hi*8 (C/D row layout)
    float ni[8];
#pragma unroll
    for (int r = 0; r < 8; ++r) {
        const int m = r + hi * 8;
        ni[r] = __shfl(sA, m, 32) + __shfl(sA, m + 16, 32);
    }

    float wacc = 0.0f;
    const int tjBeg = (tjBase > ti) ? tjBase : ti;
    const int tjEnd = (tjBase + 8 < nTiles) ? (tjBase + 8) : nTiles;
    for (int tj = tjBeg; tj < tjEnd; ++tj) {
        // --- B fragment: same per-lane formula on the column tile ---
        const int bj = tj * 16 + la;
        v2f bfrag;
        if (hi == 0) {
            bfrag[0] = geom[3 * bj + 0];
            bfrag[1] = geom[3 * bj + 1];
        } else {
            bfrag[0] = geom[3 * bj + 2];
            bfrag[1] = 0.0f;
        }
        const float sB = bfrag[0] * bfrag[0] + bfrag[1] * bfrag[1];
        const float nj = __shfl(sB, la, 32) + __shfl(sB, la + 16, 32);

        // --- dot-product tile via CDNA5 WMMA ---
        v8f c = {};
        c = __builtin_amdgcn_wmma_f32_16x16x4_f32(
                /*neg_a=*/false, afrag, /*neg_b=*/false, bfrag,
                /*c_mod=*/(short)0, c, /*reuse_a=*/false, /*reuse_b=*/false);

        const bool diag = (ti == tj);
#pragma unroll
        for (int r = 0; r < 8; ++r) {
            const int m  = r + hi * 8;              // row index within tile
            float sq = ni[r] + nj - 2.0f * c[r];
            sq = fmaxf(sq, 0.0f);                    // guard fp rounding < 0
            const float d = fast_sqrtf(sq);
            const float p = fmaxf(THRESH_MIN - d, 0.0f) +
                            fmaxf(d - THRESH_MAX, 0.0f);
            const bool valid = (!diag) || (m < la);  // i < j
            wacc += valid ? p : 0.0f;
        }
    }

    // --- wave32 butterfly reduction ---
#pragma unroll
    for (int off = 16; off > 0; off >>= 1)
        wacc += __shfl_xor(wacc, off, 32);

    if (lane == 0) {
        // deterministic fixed-point accumulate (integer atomic)
        const unsigned long long q =
            (unsigned long long)((double)wacc * FIXED_SCALE + 0.5);
        atomicAdd(acc, q);
    }
}

// ---- finalize: dequantize and normalize by pair count ----
__global__ void dp_final_kernel(const unsigned long long* __restrict__ acc,
                                float* __restrict__ out,
                                double inv_scale_pairs) {
    if (threadIdx.x == 0)
        out[0] = (float)((double)acc[0] * inv_scale_pairs);
}

extern "C" void kernel_launch(void* const* d_in, const int* in_sizes, int n_in,
                              void* d_out, int out_size, void* d_ws, size_t ws_size,
                              hipStream_t stream) {
    const float* geom = (const float*)d_in[0];
    const int n3     = in_sizes[0];
    const int N      = n3 / 3;          // 4096
    const int nTiles = N / 16;          // 256 (N is a multiple of 16)

    unsigned long long* acc = (unsigned long long*)d_ws;
    float* out = (float*)d_out;

    dp_init_kernel<<<1, 32, 0, stream>>>(acc);

    dim3 grid((nTiles + 7) / 8, nTiles);   // (32, 256): 8-tile strips x row tiles
    dp_pairs_kernel<<<grid, 32, 0, stream>>>(geom, acc, nTiles);

    const double num_pairs = (double)N * (double)(N - 1) * 0.5;
    dp_final_kernel<<<1, 32, 0, stream>>>(acc, out,
                                          1.0 / (FIXED_SCALE * num_pairs));
}